// FrequencyLayer_50500225466927
// MI455X (gfx1250) — compile-verified
//
#include <hip/hip_runtime.h>

typedef float v2f __attribute__((ext_vector_type(2)));
typedef float v8f __attribute__((ext_vector_type(8)));

#define FL_S   2048
#define FL_H   1024
#define FL_C   4
#define NSEG   8
#define SSEG   (FL_S / NSEG)       // 256
#define LN_EPS 1e-12f
#define TWO_PI_F 6.28318530717958647692f

// ---------------- K0: basis tables -------------------------------------------
// proj[s][8] = {1, cos(th1), sin(th1), cos(th2), sin(th2), cos(th3), sin(th3), 0}
// rec [s][8] = proj scaled by {1/S, 2/S, 2/S, ...}, slot 7 = 0 (K-pad for WMMA)
__global__ void fl_tables(float* __restrict__ proj, float* __restrict__ rec) {
  int s = blockIdx.x * blockDim.x + threadIdx.x;
  if (s >= FL_S) return;
  float base = (TWO_PI_F / (float)FL_S) * (float)s;
  float* p = proj + s * 8;
  float* r = rec  + s * 8;
  p[0] = 1.0f;               r[0] = 1.0f / (float)FL_S;
  #pragma unroll
  for (int k = 1; k < FL_C; ++k) {
    float c = cosf(base * (float)k);
    float sn = sinf(base * (float)k);
    p[2 * k - 1] = c;        r[2 * k - 1] = (2.0f / (float)FL_S) * c;
    p[2 * k]     = sn;       r[2 * k]     = (2.0f / (float)FL_S) * sn;
  }
  p[7] = 0.0f;               r[7] = 0.0f;
}

// ---------------- K1: projection partials ------------------------------------
// grid = B * NSEG * (H/256); partials layout [seg][b][8][H], j=7 padded zero
__global__ __launch_bounds__(256) void fl_project(
    const float* __restrict__ x, const float* __restrict__ proj,
    float* __restrict__ partials, int B) {
  const int blk   = blockIdx.x;
  const int htile = blk & 3;                 // H/256 = 4
  const int seg   = (blk >> 2) & (NSEG - 1);
  const int b     = blk >> 5;                // / (NSEG*4)
  const int h     = htile * 256 + threadIdx.x;

  __shared__ float pl[SSEG * 8];             // 8 KB: basis rows for this segment
  for (int i = threadIdx.x; i < SSEG * 8; i += 256)
    pl[i] = proj[(size_t)(seg * SSEG) * 8 + i];
  __syncthreads();

  float acc[7];
  #pragma unroll
  for (int j = 0; j < 7; ++j) acc[j] = 0.0f;

  const float* xp = x + (size_t)(b * FL_S + seg * SSEG) * FL_H + h;
  #pragma unroll 4
  for (int s = 0; s < SSEG; ++s) {
    float xv = xp[(size_t)s * FL_H];         // coalesced across threads
    const float* pr = &pl[s * 8];            // LDS broadcast
    #pragma unroll
    for (int j = 0; j < 7; ++j) acc[j] = fmaf(xv, pr[j], acc[j]);
  }
  float* pp = partials + (size_t)(seg * B + b) * 8 * FL_H + h;
  #pragma unroll
  for (int j = 0; j < 7; ++j) pp[(size_t)j * FL_H] = acc[j];
  pp[(size_t)7 * FL_H] = 0.0f;
}

// ---------------- K1b: deterministic segment reduce --------------------------
__global__ void fl_reduce(const float* __restrict__ partials,
                          float* __restrict__ coeff, int n, int strideSeg) {
  int i = blockIdx.x * blockDim.x + threadIdx.x;
  if (i >= n) return;
  float s = 0.0f;
  #pragma unroll
  for (int seg = 0; seg < NSEG; ++seg) s += partials[(size_t)seg * strideSeg + i];
  coeff[i] = s;
}

// ---------------- K2: WMMA reconstruct + mix + fused LayerNorm ---------------
// one block = one (b, 16-row s-tile); 512 threads = 16 wave32
// low[16s x 16h] = Rec[16x8] @ Cf[8x16] via two V_WMMA_F32_16X16X4_F32 (K=8)
__global__ __launch_bounds__(512) void fl_recln(
    const float* __restrict__ x, const float* __restrict__ sqrtb,
    const float* __restrict__ gamma, const float* __restrict__ beta,
    const float* __restrict__ rec, const float* __restrict__ coeff,
    float* __restrict__ out) {
  const int blk  = blockIdx.x;
  const int b    = blk >> 7;                 // S/16 = 128 s-tiles per batch
  const int s0   = (blk & 127) << 4;
  const int tid  = threadIdx.x;
  const int wave = tid >> 5;                 // 0..15
  const int lane = tid & 31;
  const int hf   = lane >> 4;                // half-wave: rows 0..7 vs 8..15
  const int ln   = lane & 15;

  __shared__ float redS[16][16];
  __shared__ float redQ[16][16];
  __shared__ float mrow[16];
  __shared__ float rrowv[16];

  // A-matrix (ISA 16x4 f32 layout): lane<16 -> M=ln,K={0,1}; lane>=16 -> K={2,3}
  const float* rr = rec + (size_t)(s0 + ln) * 8 + hf * 2;
  v2f a0, a1;
  a0.x = rr[0]; a0.y = rr[1];                // j = 0..3 slice
  a1.x = rr[4]; a1.y = rr[5];                // j = 4..7 slice (j=7 is zero)

  const float* cbase = coeff + (size_t)(b * 8) * FL_H;
  const float* xbase = x   + (size_t)(b * FL_S + s0 + hf * 8) * FL_H;
  float*       obase = out + (size_t)(b * FL_S + s0 + hf * 8) * FL_H;

  float tt[4][8];
  float rs[8], rq[8];
  #pragma unroll
  for (int r = 0; r < 8; ++r) { rs[r] = 0.0f; rq[r] = 0.0f; }

  #pragma unroll
  for (int it = 0; it < 4; ++it) {           // 4 h-tiles of 16 per wave
    const int h = (wave * 4 + it) * 16 + ln;
    const float sb  = sqrtb[h];
    const float sb2 = sb * sb;
    // B-matrix (4x16): half0 supplies K rows {0,1}, half1 rows {2,3}
    const float* cp = cbase + h;
    v2f b0, b1;
    b0.x = cp[(size_t)(hf * 2 + 0) * FL_H];
    b0.y = cp[(size_t)(hf * 2 + 1) * FL_H];
    b1.x = cp[(size_t)(hf * 2 + 4) * FL_H];
    b1.y = cp[(size_t)(hf * 2 + 5) * FL_H];
    v8f c = {};
    c = __builtin_amdgcn_wmma_f32_16x16x4_f32(false, a0, false, b0, (short)0, c, false, false);
    c = __builtin_amdgcn_wmma_f32_16x16x4_f32(false, a1, false, b1, (short)0, c, false, false);
    // c[r] = low at (s = s0 + hf*8 + r, h); mix with x and accumulate LN sums
    const float* xp = xbase + h;
    #pragma unroll
    for (int r = 0; r < 8; ++r) {
      float xv = xp[(size_t)r * FL_H];
      float t  = (1.0f - sb2) * c[r] + (1.0f + sb2) * xv;  // low + b2*(x-low) + x
      tt[it][r] = t;
      rs[r] += t;
      rq[r] += t * t;
    }
  }

  // reduce over the 16 lanes of each half-wave (xor <= 8 stays inside a half)
  #pragma unroll
  for (int r = 0; r < 8; ++r) {
    #pragma unroll
    for (int off = 1; off <= 8; off <<= 1) {
      rs[r] += __shfl_xor(rs[r], off, 32);
      rq[r] += __shfl_xor(rq[r], off, 32);
    }
  }
  if (ln == 0) {
    #pragma unroll
    for (int r = 0; r < 8; ++r) {
      redS[wave][hf * 8 + r] = rs[r];
      redQ[wave][hf * 8 + r] = rq[r];
    }
  }
  __syncthreads();
  if (tid < 16) {                             // fixed-order cross-wave reduce
    float ssum = 0.0f, qsum = 0.0f;
    #pragma unroll
    for (int w = 0; w < 16; ++w) { ssum += redS[w][tid]; qsum += redQ[w][tid]; }
    float mean = ssum * (1.0f / (float)FL_H);
    float var  = qsum * (1.0f / (float)FL_H) - mean * mean;
    mrow[tid]  = mean;
    rrowv[tid] = rsqrtf(var + LN_EPS);
  }
  __syncthreads();

  #pragma unroll
  for (int it = 0; it < 4; ++it) {
    const int h = (wave * 4 + it) * 16 + ln;
    const float g  = gamma[h];
    const float bb = beta[h];
    float* op = obase + h;
    #pragma unroll
    for (int r = 0; r < 8; ++r) {
      float mean = mrow[hf * 8 + r];
      float rstd = rrowv[hf * 8 + r];
      op[(size_t)r * FL_H] = (tt[it][r] - mean) * rstd * g + bb;
    }
  }
}

// ---------------- launcher ---------------------------------------------------
extern "C" void kernel_launch(void* const* d_in, const int* in_sizes, int n_in,
                              void* d_out, int out_size, void* d_ws, size_t ws_size,
                              hipStream_t stream) {
  const float* x     = (const float*)d_in[0];
  const float* sqrtb = (const float*)d_in[1];
  const float* gamma = (const float*)d_in[2];
  const float* beta  = (const float*)d_in[3];
  float*       out   = (float*)d_out;
  const int B = in_sizes[0] / (FL_S * FL_H);

  float* ws       = (float*)d_ws;
  float* proj     = ws;                               // S*8 floats
  float* rec      = proj + (size_t)FL_S * 8;          // S*8 floats
  float* coeff    = rec  + (size_t)FL_S * 8;          // B*8*H floats
  float* partials = coeff + (size_t)B * 8 * FL_H;     // NSEG*B*8*H floats

  fl_tables<<<(FL_S + 255) / 256, 256, 0, stream>>>(proj, rec);
  fl_project<<<B * NSEG * (FL_H / 256), 256, 0, stream>>>(x, proj, partials, B);
  const int n = B * 8 * FL_H;
  fl_reduce<<<(n + 255) / 256, 256, 0, stream>>>(partials, coeff, n, n);
  fl_recln<<<B * (FL_S / 16), 512, 0, stream>>>(x, sqrtb, gamma, beta, rec, coeff, out);
}